// Mesh_59931973648984
// MI455X (gfx1250) — compile-verified
//
#include <hip/hip_runtime.h>
#include <math.h>

// ---------------------------------------------------------------------------
// Problem constants (match reference)
// ---------------------------------------------------------------------------
#define H_IMG 160
#define W_IMG 160
#define NPIX  (H_IMG * W_IMG)     // 25600
#define NV    8192
#define NF    512
#define NC    16
#define NCH   (NC + 1)            // 17 output channels
#define NQUAD (NF / 4)            // 128 triangle quads (4 tris per WMMA)
#define SIGMA 0.5f

typedef __attribute__((ext_vector_type(2))) float v2f;
typedef __attribute__((ext_vector_type(8))) float v8f;

// D(16x16) = A(16x4) x B(4x16) + C   — the only f32 WMMA shape on CDNA5.
__device__ inline v8f wmma_f32_16x16x4(v2f a, v2f b) {
  v8f c = {0.f, 0.f, 0.f, 0.f, 0.f, 0.f, 0.f, 0.f};
  return __builtin_amdgcn_wmma_f32_16x16x4_f32(
      /*neg_a=*/false, a, /*neg_b=*/false, b,
      /*c_mod=*/(short)0, c, /*reuse_a=*/false, /*reuse_b=*/false);
}

__device__ inline float fast_rcp(float x) {
#if __has_builtin(__builtin_amdgcn_rcpf)
  return __builtin_amdgcn_rcpf(x);   // v_rcp_f32, ~1 ulp — plenty for argmin key
#else
  return 1.0f / x;
#endif
}

// ---------------------------------------------------------------------------
// Kernel 0: zero accumulators (d_out is poisoned by the harness)
// ---------------------------------------------------------------------------
__global__ void k_zero(float* __restrict__ out, float* __restrict__ norm) {
  int i = blockIdx.x * blockDim.x + threadIdx.x;
  if (i < NPIX * NCH)            out[i] = 0.f;
  else if (i < NPIX * (NCH + 1)) norm[i - NPIX * NCH] = 0.f;
}

// ---------------------------------------------------------------------------
// Kernel 1: vertex transform  clip = [x,y,z,1] @ M^T
// ---------------------------------------------------------------------------
__global__ void k_vertex(const float* __restrict__ vtx,
                         const float* __restrict__ m,
                         float* __restrict__ clip) {
  int v = blockIdx.x * blockDim.x + threadIdx.x;
  if (v >= NV) return;
  float x = vtx[v * 3 + 0], y = vtx[v * 3 + 1], z = vtx[v * 3 + 2];
#pragma unroll
  for (int i = 0; i < 4; ++i)
    clip[v * 4 + i] = m[i * 4 + 0] * x + m[i * 4 + 1] * y + m[i * 4 + 2] * z + m[i * 4 + 3];
}

// ---------------------------------------------------------------------------
// Kernel 2: per-triangle affine-coefficient setup.
// 4 quantities per triangle, each affine in (px,py): f = cx*px + cy*py + cc
//   q0..q2 : bw_k = (e_k / area_safe) / w_k     (sign(bw_k)==sign(b_k) since w>0)
//   q3     : depth numerator  sum_k bw_k * z_k
// A-matrix rows: q = (t%4)*4 + local_q  → 4 triangles per 16-row WMMA tile.
// Packed for direct WMMA-A lane loads:
//   coef[quad*32 + q]      = (cx, cy)   (lanes 0..15  hold A K=0,1)
//   coef[quad*32 + 16 + q] = (cc, 0 )   (lanes 16..31 hold A K=2,3)
// ---------------------------------------------------------------------------
__global__ void k_setup(const float* __restrict__ clip,
                        const int* __restrict__ faces,
                        float2* __restrict__ coef) {
  int t = blockIdx.x * blockDim.x + threadIdx.x;
  if (t >= NF) return;
  int i0 = faces[t * 3 + 0], i1 = faces[t * 3 + 1], i2 = faces[t * 3 + 2];

  float w0 = clip[i0 * 4 + 3], w1 = clip[i1 * 4 + 3], w2 = clip[i2 * 4 + 3];
  float p0x = clip[i0 * 4 + 0] / w0, p0y = clip[i0 * 4 + 1] / w0, z0 = clip[i0 * 4 + 2] / w0;
  float p1x = clip[i1 * 4 + 0] / w1, p1y = clip[i1 * 4 + 1] / w1, z1 = clip[i1 * 4 + 2] / w1;
  float p2x = clip[i2 * 4 + 0] / w2, p2y = clip[i2 * 4 + 1] / w2, z2 = clip[i2 * 4 + 2] / w2;

  float area = (p1x - p0x) * (p2y - p0y) - (p1y - p0y) * (p2x - p0x);
  bool valid = (fabsf(area) > 1e-9f) && (w0 > 1e-6f) && (w1 > 1e-6f) && (w2 > 1e-6f);
  float area_safe = (fabsf(area) > 1e-9f) ? area : 1.0f;
  float inv_area = 1.0f / area_safe;

  // edge(a,b,q) = (bx-ax)(qy-ay) - (by-ay)(qx-ax)
  //   cx = -(by-ay), cy = (bx-ax), cc = (by-ay)*ax - (bx-ax)*ay
  float cx[4], cy[4], cc[4];
  float s0 = inv_area / w0, s1 = inv_area / w1, s2 = inv_area / w2;
  // e0: a=p1,b=p2 ; e1: a=p2,b=p0 ; e2: a=p0,b=p1   (scaled by 1/(area*w_k))
  cx[0] = -(p2y - p1y) * s0; cy[0] = (p2x - p1x) * s0;
  cc[0] = ((p2y - p1y) * p1x - (p2x - p1x) * p1y) * s0;
  cx[1] = -(p0y - p2y) * s1; cy[1] = (p0x - p2x) * s1;
  cc[1] = ((p0y - p2y) * p2x - (p0x - p2x) * p2y) * s1;
  cx[2] = -(p1y - p0y) * s2; cy[2] = (p1x - p0x) * s2;
  cc[2] = ((p1y - p0y) * p0x - (p1x - p0x) * p0y) * s2;
  // q3: depth numerator = sum_k bw_k * z_k
  cx[3] = cx[0] * z0 + cx[1] * z1 + cx[2] * z2;
  cy[3] = cy[0] * z0 + cy[1] * z1 + cy[2] * z2;
  cc[3] = cc[0] * z0 + cc[1] * z1 + cc[2] * z2;

  if (!valid) {
#pragma unroll
    for (int q = 0; q < 4; ++q) { cx[q] = 0.f; cy[q] = 0.f; cc[q] = 0.f; }
    cc[0] = -1.0f;  // bw0 == -1 everywhere -> never passes inside test
  }

  int quad = t >> 2;
  int qb = (t & 3) * 4;
#pragma unroll
  for (int q = 0; q < 4; ++q) {
    coef[quad * 32 + qb + q]      = make_float2(cx[q], cy[q]);
    coef[quad * 32 + 16 + qb + q] = make_float2(cc[q], 0.0f);
  }
}

// ---------------------------------------------------------------------------
// Kernel 3: rasterize.  256 threads = 8 waves; wave handles one 16-pixel tile.
// Per WMMA: 16 pixels x 4 triangles; each lane owns two (pixel,tri) pairs in
// its 8 accumulator VGPRs -> argmin tracking is pure per-lane VALU.
// ---------------------------------------------------------------------------
__global__ void __launch_bounds__(256) k_raster(const float2* __restrict__ coef,
                                                float4* __restrict__ outBary,
                                                int* __restrict__ outId) {
  __shared__ float2 sc[NQUAD * 32];  // 32 KB of LDS, shared by all 8 waves
  int tid = threadIdx.x;
  for (int i = tid; i < NQUAD * 32; i += 256) sc[i] = coef[i];
  __syncthreads();

  int lane = tid & 31;
  int wave = tid >> 5;
  int tile = blockIdx.x * 8 + wave;          // 200 blocks * 8 = 1600 tiles
  int n = lane & 15;
  int pix = tile * 16 + n;                   // W=160 is a multiple of 16
  int row = pix / W_IMG, colp = pix % W_IMG;
  float px = ((float)colp + 0.5f) / (float)W_IMG * 2.0f - 1.0f;
  float py = ((float)row + 0.5f) / (float)H_IMG * 2.0f - 1.0f;

  // B(4x16): rows K0=px, K1=py, K2=1, K3=0  (K = vgpr + 2*(lane>=16))
  v2f bmat;
  bmat.x = (lane < 16) ? px : 1.0f;
  bmat.y = (lane < 16) ? py : 0.0f;

  // This lane's triangle ids per quad: base + {0,1} (lanes<16) or base + {2,3}.
  int subBase = (lane >> 4) * 2;

  float bestD = __builtin_inff();
  int bestId = 0;
  float bw0b = 0.f, bw1b = 0.f, bw2b = 0.f, Sb = 1.0f;

  float2 cf = sc[lane];  // software-pipelined coefficient fetch
#pragma unroll 4
  for (int p = 0; p < NQUAD; ++p) {
    v2f a; a.x = cf.x; a.y = cf.y;
    cf = sc[(((p + 1) & (NQUAD - 1)) * 32) + lane];  // prefetch next quad
    v8f d = wmma_f32_16x16x4(a, bmat);

#pragma unroll
    for (int half = 0; half < 2; ++half) {           // ascending tri id order
      float bw0 = d[half * 4 + 0], bw1 = d[half * 4 + 1];
      float bw2 = d[half * 4 + 2], nz = d[half * 4 + 3];
      bool inside = (bw0 >= 0.f) && (bw1 >= 0.f) && (bw2 >= 0.f);
      float S = bw0 + bw1 + bw2;
      float Ss = (fabsf(S) > 1e-12f) ? S : 1.0f;
      float depth = inside ? (nz * fast_rcp(Ss)) : __builtin_inff();
      bool better = depth < bestD;  // strict < keeps first (argmin) semantics
      bestId = better ? (4 * p + subBase + half) : bestId;
      bw0b = better ? bw0 : bw0b;
      bw1b = better ? bw1 : bw1b;
      bw2b = better ? bw2 : bw2b;
      Sb   = better ? Ss : Sb;
      bestD = better ? depth : bestD;
    }
  }

  // perspective-correct barycentrics of this lane's winner (precise, once)
  float inv = 1.0f / Sb;
  float c0 = bw0b * inv, c1 = bw1b * inv, c2 = bw2b * inv;

  // merge tri-{0,1 mod 4} (lanes 0..15) with tri-{2,3 mod 4} (lanes 16..31)
  float dH = __shfl_xor(bestD, 16, 32);
  int   iH = __shfl_xor(bestId, 16, 32);
  float h0 = __shfl_xor(c0, 16, 32);
  float h1 = __shfl_xor(c1, 16, 32);
  float h2 = __shfl_xor(c2, 16, 32);
  if (lane < 16) {
    bool hi = (dH < bestD) || ((dH == bestD) && (iH < bestId));
    float dsel = hi ? dH : bestD;
    int id = hi ? iH : bestId;
    float b0 = hi ? h0 : c0, b1 = hi ? h1 : c1, b2 = hi ? h2 : c2;
    float mf = (dsel < 1e30f) ? 1.0f : 0.0f;   // mask = any triangle covered
    outBary[pix] = make_float4(b0 * mf, b1 * mf, b2 * mf, mf);
    outId[pix] = id;
  }
}

// ---------------------------------------------------------------------------
// Kernel 4: attribute/clip interpolation + Gaussian 3x3 splat (scatter-add)
// ---------------------------------------------------------------------------
__global__ void k_splat(const float4* __restrict__ bary,
                        const int* __restrict__ ids,
                        const int* __restrict__ faces,
                        const float* __restrict__ clip,
                        const float* __restrict__ attr,
                        float* __restrict__ accum,
                        float* __restrict__ norm) {
  int p = blockIdx.x * blockDim.x + threadIdx.x;
  if (p >= NPIX) return;
  float4 b = bary[p];
  if (b.w == 0.f) return;  // masked pixels contribute exactly zero
  int t = ids[p];
  int v0 = faces[t * 3 + 0], v1 = faces[t * 3 + 1], v2 = faces[t * 3 + 2];

  float col[NCH];
#pragma unroll
  for (int c = 0; c < NC; ++c)
    col[c] = b.x * attr[v0 * NC + c] + b.y * attr[v1 * NC + c] + b.z * attr[v2 * NC + c];
  col[NC] = 1.0f;  // maskf channel

  float cb[4];
#pragma unroll
  for (int c = 0; c < 4; ++c)
    cb[c] = b.x * clip[v0 * 4 + c] + b.y * clip[v1 * 4 + c] + b.z * clip[v2 * 4 + c];
  float iw = 1.0f / (cb[3] + 1e-6f);
  float vx = (cb[0] * iw + 1.0f) * (0.5f * (float)W_IMG);
  float vy = (cb[1] * iw + 1.0f) * (0.5f * (float)H_IMG);
  float bj = floorf(vx), bi = floorf(vy);
  const float inv2s2 = 0.5f / (SIGMA * SIGMA);

#pragma unroll
  for (int dy = -1; dy <= 1; ++dy) {
#pragma unroll
    for (int dx = -1; dx <= 1; ++dx) {
      float tj = bj + (float)dx, ti = bi + (float)dy;
      if (tj < 0.f || tj >= (float)W_IMG || ti < 0.f || ti >= (float)H_IMG) continue;
      float ddx = vx - (tj + 0.5f), ddy = vy - (ti + 0.5f);
      float w = expf(-(ddx * ddx + ddy * ddy) * inv2s2);
      int idx = (int)ti * W_IMG + (int)tj;
#pragma unroll
      for (int c = 0; c < NCH; ++c) atomicAdd(&accum[idx * NCH + c], w * col[c]);
      atomicAdd(&norm[idx], w);
    }
  }
}

// ---------------------------------------------------------------------------
// Kernel 5: normalize
// ---------------------------------------------------------------------------
__global__ void k_norm(float* __restrict__ out, const float* __restrict__ norm) {
  int i = blockIdx.x * blockDim.x + threadIdx.x;
  if (i >= NPIX * NCH) return;
  out[i] = out[i] / (norm[i / NCH] + 1e-8f);
}

// ---------------------------------------------------------------------------
// Host-side launcher
// ---------------------------------------------------------------------------
extern "C" void kernel_launch(void* const* d_in, const int* in_sizes, int n_in,
                              void* d_out, int out_size, void* d_ws, size_t ws_size,
                              hipStream_t stream) {
  const float* vtx   = (const float*)d_in[0];   // V*3
  const int*   faces = (const int*)d_in[1];     // F*3
  const float* w2c   = (const float*)d_in[2];   // 16
  const float* attr  = (const float*)d_in[3];   // V*16
  float* out = (float*)d_out;                   // NPIX*17

  char* ws = (char*)d_ws;
  float*  clip  = (float*)(ws);                              // NV*4 f32   (131072 B)
  float2* coef  = (float2*)(ws + 131072);                    // NQUAD*32   ( 32768 B)
  float4* baryb = (float4*)(ws + 196608);                    // NPIX       (409600 B)
  int*    idsb  = (int*)(ws + 606208);                       // NPIX       (102400 B)
  float*  normb = (float*)(ws + 708608);                     // NPIX       (102400 B)

  k_zero<<<(NPIX * (NCH + 1) + 255) / 256, 256, 0, stream>>>(out, normb);
  k_vertex<<<(NV + 255) / 256, 256, 0, stream>>>(vtx, w2c, clip);
  k_setup<<<(NF + 255) / 256, 256, 0, stream>>>(clip, faces, coef);
  k_raster<<<(NPIX / 16) / 8, 256, 0, stream>>>(coef, baryb, idsb);
  k_splat<<<(NPIX + 255) / 256, 256, 0, stream>>>(baryb, idsb, faces, clip, attr, out, normb);
  k_norm<<<(NPIX * NCH + 255) / 256, 256, 0, stream>>>(out, normb);
}